// SNAP_GNN_LITE_34840774705775
// MI455X (gfx1250) — compile-verified
//
#include <hip/hip_runtime.h>
#include <hip/hip_bf16.h>
#include <cmath>

// -------- WMMA vector types (f32 16x16x4: A/B = 2 VGPRs/lane, C/D = 8) -----
typedef __attribute__((ext_vector_type(2))) float v2f;
typedef __attribute__((ext_vector_type(8))) float v8f;

#define IN_DIM 256
#define LATENT 32

// ---------------------------------------------------------------------------
// Degree (with self loop) -> dis = rsqrt(deg)
// ---------------------------------------------------------------------------
__global__ void deg_init_kernel(unsigned* deg, int n) {
  int i = blockIdx.x * blockDim.x + threadIdx.x;
  if (i < n) deg[i] = 1u;  // self loop contributes 1
}

__global__ void deg_count_kernel(const int* __restrict__ dst, unsigned* deg, int e) {
  int i = blockIdx.x * blockDim.x + threadIdx.x;
  if (i < e) atomicAdd(&deg[dst[i]], 1u);  // global_atomic_add_u32
}

__global__ void deg_final_kernel(const unsigned* __restrict__ deg, float* dis, int n) {
  int i = blockIdx.x * blockDim.x + threadIdx.x;
  if (i < n) dis[i] = rsqrtf((float)deg[i]);
}

// ---------------------------------------------------------------------------
// h0 = relu(x @ fc_w + fc_b)   [N,256] x [256,32]  via v_wmma_f32_16x16x4_f32
// One wave computes a 16-row x 32-col tile (two 16x16 accumulators).
// A fragment (16x4 f32): lane l -> row = l&15, k-pair = (l>>4)*2
// B fragment (4x16 f32): lane l -> col = l&15, k-pair = (l>>4)*2
// C/D (16x16 f32): vgpr r -> row = (l>>4)*8 + r, col = l&15
// ---------------------------------------------------------------------------
__global__ void fc_relu_wmma_kernel(const float* __restrict__ x,
                                    const float* __restrict__ w,     // [256,32]
                                    const float* __restrict__ bias,  // [32]
                                    float* __restrict__ h, int n) {
  int wave = (int)((blockIdx.x * (long long)blockDim.x + threadIdx.x) >> 5);
  int lane = threadIdx.x & 31;
  int m0 = wave * 16;
  if (m0 >= n) return;  // uniform per wave -> whole wave exits, WMMA EXEC stays all-1s
  int r16 = lane & 15;
  int kh  = (lane >> 4) << 1;  // 0 or 2
  const float* xrow = x + (size_t)(m0 + r16) * IN_DIM;
  v8f acc0 = {};
  v8f acc1 = {};
  for (int k = 0; k < IN_DIM; k += 4) {
    v2f a, b0, b1;
    a[0] = xrow[k + kh];
    a[1] = xrow[k + kh + 1];
    const float* wr = w + (size_t)(k + kh) * LATENT;
    b0[0] = wr[r16];           b0[1] = wr[LATENT + r16];
    b1[0] = wr[16 + r16];      b1[1] = wr[LATENT + 16 + r16];
    acc0 = __builtin_amdgcn_wmma_f32_16x16x4_f32(false, a, false, b0, (short)0, acc0, false, false);
    acc1 = __builtin_amdgcn_wmma_f32_16x16x4_f32(false, a, false, b1, (short)0, acc1, false, false);
  }
  float bb0 = bias[r16], bb1 = bias[16 + r16];
  int rbase = m0 + ((lane >> 4) << 3);
  for (int r = 0; r < 8; ++r) {
    float v0 = fmaxf(acc0[r] + bb0, 0.f);
    float v1 = fmaxf(acc1[r] + bb1, 0.f);
    float* orow = h + (size_t)(rbase + r) * LATENT;
    orow[r16]      = v0;
    orow[16 + r16] = v1;
  }
}

// ---------------------------------------------------------------------------
// g = (RELU_IN ? relu(hin) : hin) @ w    [N,32] x [32,32]  via WMMA f32
// ---------------------------------------------------------------------------
template <bool RELU_IN>
__global__ void gemm32_wmma_kernel(const float* __restrict__ hin,
                                   const float* __restrict__ w,   // [32,32]
                                   float* __restrict__ gout, int n) {
  int wave = (int)((blockIdx.x * (long long)blockDim.x + threadIdx.x) >> 5);
  int lane = threadIdx.x & 31;
  int m0 = wave * 16;
  if (m0 >= n) return;
  int r16 = lane & 15;
  int kh  = (lane >> 4) << 1;
  const float* hrow = hin + (size_t)(m0 + r16) * LATENT;
  v8f acc0 = {};
  v8f acc1 = {};
#pragma unroll
  for (int k = 0; k < LATENT; k += 4) {
    float a0 = hrow[k + kh];
    float a1 = hrow[k + kh + 1];
    if (RELU_IN) { a0 = fmaxf(a0, 0.f); a1 = fmaxf(a1, 0.f); }
    v2f a, b0, b1;
    a[0] = a0; a[1] = a1;
    const float* wr = w + (size_t)(k + kh) * LATENT;
    b0[0] = wr[r16];           b0[1] = wr[LATENT + r16];
    b1[0] = wr[16 + r16];      b1[1] = wr[LATENT + 16 + r16];
    acc0 = __builtin_amdgcn_wmma_f32_16x16x4_f32(false, a, false, b0, (short)0, acc0, false, false);
    acc1 = __builtin_amdgcn_wmma_f32_16x16x4_f32(false, a, false, b1, (short)0, acc1, false, false);
  }
  int rbase = m0 + ((lane >> 4) << 3);
  for (int r = 0; r < 8; ++r) {
    float* orow = gout + (size_t)(rbase + r) * LATENT;
    orow[r16]      = acc0[r];
    orow[16 + r16] = acc1[r];
  }
}

// ---------------------------------------------------------------------------
// out[i,f] = bias[f] + dis[i]^2 * g[i,f]   (self-loop term + bias, full init)
// ---------------------------------------------------------------------------
__global__ void conv_init_kernel(const float* __restrict__ g,
                                 const float* __restrict__ dis,
                                 const float* __restrict__ bias,
                                 float* __restrict__ out, int n) {
  int i = blockIdx.x * blockDim.x + threadIdx.x;
  int node = i >> 5;
  if (node >= n) return;
  int f = i & 31;
  float d = dis[node];
  out[i] = bias[f] + d * d * g[i];
}

// ---------------------------------------------------------------------------
// Edge scatter: wave per edge, lane = feature.
// out[dst] += g[src] * dis[src] * dis[dst]   (L2-resident float atomics)
// ---------------------------------------------------------------------------
__global__ void conv_scatter_kernel(const float* __restrict__ g,
                                    const float* __restrict__ dis,
                                    const int* __restrict__ src,
                                    const int* __restrict__ dst,
                                    float* out, int e) {
  int lane = threadIdx.x & 31;
  long long wid    = (((long long)blockIdx.x * blockDim.x) + threadIdx.x) >> 5;
  long long stride = ((long long)gridDim.x * blockDim.x) >> 5;
  for (long long i = wid; i < e; i += stride) {
    int s = src[i];
    int d = dst[i];
    float wgt = dis[s] * dis[d];
    float val = g[(size_t)s * LATENT + lane] * wgt;
    unsafeAtomicAdd(&out[(size_t)d * LATENT + lane], val);  // global_atomic_add_f32
  }
}

// ---------------------------------------------------------------------------
extern "C" void kernel_launch(void* const* d_in, const int* in_sizes, int n_in,
                              void* d_out, int out_size, void* d_ws, size_t ws_size,
                              hipStream_t stream) {
  const float* x    = (const float*)d_in[0];
  const int*   eidx = (const int*)d_in[1];
  const float* fc_w = (const float*)d_in[2];
  const float* fc_b = (const float*)d_in[3];
  const float* w1   = (const float*)d_in[4];
  const float* b1   = (const float*)d_in[5];
  const float* w2   = (const float*)d_in[6];
  const float* b2   = (const float*)d_in[7];

  const int n = in_sizes[0] / IN_DIM;  // 100000
  const int e = in_sizes[1] / 2;       // 3200000
  const int* src = eidx;
  const int* dst = eidx + e;

  // workspace layout
  char* ws = (char*)d_ws;
  unsigned* deg = (unsigned*)ws;                       // n u32
  float* dis = (float*)(ws + (size_t)n * 4);           // n f32
  float* h0  = (float*)(ws + (size_t)n * 8);           // n*32 f32
  float* g   = h0 + (size_t)n * LATENT;                // n*32 f32
  float* h1  = g + (size_t)n * LATENT;                 // n*32 f32
  float* out = (float*)d_out;                          // n*32 f32

  const int tb = 256;  // 8 wave32's per block
  const int nb_nodes = (n + tb - 1) / tb;
  const int nb_edges = (e + tb - 1) / tb;
  const int nwaves   = (n + 15) / 16;
  const int nb_gemm  = (nwaves + 7) / 8;
  const int nb_feat  = ((long long)n * LATENT + tb - 1) / tb;
  const long long sc_threads = (long long)e * 32;
  const int nb_scatter = (int)((sc_threads + tb - 1) / tb);

  // symmetric normalization coefficients (shared by both convs)
  deg_init_kernel<<<nb_nodes, tb, 0, stream>>>(deg, n);
  deg_count_kernel<<<nb_edges, tb, 0, stream>>>(dst, deg, e);
  deg_final_kernel<<<nb_nodes, tb, 0, stream>>>(deg, dis, n);

  // h0 = relu(x @ fc_w + fc_b)
  fc_relu_wmma_kernel<<<nb_gemm, tb, 0, stream>>>(x, fc_w, fc_b, h0, n);

  // conv1: g = h0 @ w1 ; h1 = scatter(g) + self + b1
  gemm32_wmma_kernel<false><<<nb_gemm, tb, 0, stream>>>(h0, w1, g, n);
  conv_init_kernel<<<nb_feat, tb, 0, stream>>>(g, dis, b1, h1, n);
  conv_scatter_kernel<<<nb_scatter, tb, 0, stream>>>(g, dis, src, dst, h1, e);

  // conv2 (relu folded into A-fragment load): g = relu(h1) @ w2
  gemm32_wmma_kernel<true><<<nb_gemm, tb, 0, stream>>>(h1, w2, g, n);
  conv_init_kernel<<<nb_feat, tb, 0, stream>>>(g, dis, b2, out, n);
  conv_scatter_kernel<<<nb_scatter, tb, 0, stream>>>(g, dis, src, dst, out, e);
}